// Rank_IGR_Loss_22316650070597
// MI455X (gfx1250) — compile-verified
//
#include <hip/hip_runtime.h>

typedef __attribute__((ext_vector_type(2))) float v2f;
typedef __attribute__((ext_vector_type(8))) float v8f;

#define NANCH 1024
#define NTHREADS 1024
#define NEGK (-1.0e30f)
#define GAMMA 3.0f

// ---------------------------------------------------------------------------
// Per-pass rank loss:
//   inputs: s_key/s_val filled per-thread (key = sort key, val = payload)
//   output (valid on tid==0 only): sum_{i<j<np} exp(-GAMMA*(v_i - v_j))
//   where v is the payload sorted by key descending.
// Factorized: u_i = exp(-G*(v_i - m)) (i < np-1), w_j = exp(+G*(v_j - m)) (j < np)
//   loss = sum_j w_j * prefix_u(j)
// Intra-chunk (16) prefix computed via V_WMMA_F32_16X16X4_F32 with a strictly
// lower-triangular ones A-matrix; cross-chunk part via a 64-element scalar scan.
// ---------------------------------------------------------------------------
__device__ float pair_loss_pass(float* s_key, float* s_val, float* s_u, float* s_v,
                                float* s_usum, float* s_vsum, float* s_part,
                                float* s_wred, int np, int tid) {
  const int lane = tid & 31;
  const int wave = tid >> 5;

  // ---- bitonic sort (descending by key), payload carried alongside --------
  for (int k = 2; k <= NANCH; k <<= 1) {
    for (int j = k >> 1; j > 0; j >>= 1) {
      __syncthreads();
      int ixj = tid ^ j;
      if (ixj > tid) {
        float ka = s_key[tid], kb = s_key[ixj];
        bool desc = ((tid & k) == 0);
        bool doswap = desc ? (ka < kb) : (ka > kb);
        if (doswap) {
          s_key[tid] = kb; s_key[ixj] = ka;
          float va = s_val[tid], vb = s_val[ixj];
          s_val[tid] = vb; s_val[ixj] = va;
        }
      }
    }
  }
  __syncthreads();

  // ---- m = max payload over first np entries (overflow shift) -------------
  float mv = (tid < np) ? s_val[tid] : -3.0e38f;
  #pragma unroll
  for (int off = 16; off; off >>= 1) mv = fmaxf(mv, __shfl_xor(mv, off, 32));
  if (lane == 0) s_wred[wave] = mv;
  __syncthreads();
  if (tid == 0) {
    float mm = s_wred[0];
    for (int w = 1; w < 32; ++w) mm = fmaxf(mm, s_wred[w]);
    s_wred[0] = mm;
  }
  __syncthreads();
  const float m = s_wred[0];

  // ---- build u / v ---------------------------------------------------------
  {
    float pv = s_val[tid];
    s_u[tid] = (tid < np - 1) ? expf(-GAMMA * (pv - m)) : 0.0f;
    s_v[tid] = (tid < np)     ? expf( GAMMA * (pv - m)) : 0.0f;
  }
  if (tid < 4) s_part[tid] = 0.0f;
  __syncthreads();

  // ---- intra-chunk triangular pairs via WMMA (waves 0..3, full EXEC) ------
  if (wave < 4) {
    const int half  = lane >> 4;      // lane half selects K pair
    const int n     = lane & 15;      // column within tile / row M for A
    const int cbase = wave * 16;      // this wave's 16 chunk-columns
    v8f acc = {};
    #pragma unroll
    for (int q = 0; q < 4; ++q) {
      const int k0 = 4 * q + 2 * half;   // K index held in VGPR0
      const int k1 = k0 + 1;             // K index held in VGPR1
      v2f a, b;
      // A = strictly-lower-triangular ones: A[m,k] = (k < m)
      a.x = (k0 < n) ? 1.0f : 0.0f;
      a.y = (k1 < n) ? 1.0f : 0.0f;
      // B[k, col] = u[(cbase+col)*16 + k]
      b.x = s_u[(cbase + n) * 16 + k0];
      b.y = s_u[(cbase + n) * 16 + k1];
      acc = __builtin_amdgcn_wmma_f32_16x16x4_f32(
          false, a, false, b, (short)0, acc, false, false);
    }
    // D[m, col]: m = vgpr + 8*half, col = cbase + n.  lossA += v[col*16+m]*D
    float part = 0.0f;
    #pragma unroll
    for (int vv = 0; vv < 8; ++vv) {
      const int mrow = vv + 8 * half;
      part += acc[vv] * s_v[(cbase + n) * 16 + mrow];
    }
    #pragma unroll
    for (int off = 16; off; off >>= 1) part += __shfl_xor(part, off, 32);
    if (lane == 0) s_part[wave] = part;
  }
  __syncthreads();

  // ---- chunk sums for cross-chunk pairs -----------------------------------
  if (tid < 64) {
    float us = 0.0f, vs = 0.0f;
    #pragma unroll
    for (int r = 0; r < 16; ++r) {
      us += s_u[tid * 16 + r];
      vs += s_v[tid * 16 + r];
    }
    s_usum[tid] = us;
    s_vsum[tid] = vs;
  }
  __syncthreads();

  float total = 0.0f;
  if (tid == 0) {
    float run = 0.0f, lb = 0.0f;
    for (int c = 0; c < 64; ++c) { lb += s_vsum[c] * run; run += s_usum[c]; }
    total = lb + s_part[0] + s_part[1] + s_part[2] + s_part[3];
  }
  return total;  // meaningful only on tid==0
}

__global__ __launch_bounds__(NTHREADS, 1)
void rank_igr_main(const float* __restrict__ cls,
                   const int* __restrict__ label_cls,
                   const float* __restrict__ pred_loc,
                   const float* __restrict__ label_loc,
                   float* __restrict__ ws) {
  __shared__ float s_key[NANCH];
  __shared__ float s_val[NANCH];
  __shared__ float s_u[NANCH];
  __shared__ float s_v[NANCH];
  __shared__ float s_usum[64];
  __shared__ float s_vsum[64];
  __shared__ float s_part[4];
  __shared__ float s_wred[32];
  __shared__ int   s_np;

  const int b   = blockIdx.x;
  const int tid = threadIdx.x;
  const int lane = tid & 31;
  const int i = tid;                 // anchor index
  const int r = i >> 5, c = i & 31;  // HW = 32

  if (tid == 0) s_np = 0;

  // ---- per-anchor: prob, IoU, valid ---------------------------------------
  const float px = (float)(c - 16) * 8.0f;  // STRIDE = 8
  const float py = (float)(r - 16) * 8.0f;
  const float* pl = pred_loc  + (size_t)b * 4 * NANCH;
  const float* ll = label_loc + (size_t)b * 4 * NANCH;
  const float pax1 = px - pl[i],            pay1 = py - pl[NANCH + i];
  const float pax2 = px + pl[2 * NANCH + i], pay2 = py + pl[3 * NANCH + i];
  const float lax1 = px - ll[i],            lay1 = py - ll[NANCH + i];
  const float lax2 = px + ll[2 * NANCH + i], lay2 = py + ll[3 * NANCH + i];
  const float ltx = fmaxf(pax1, lax1), lty = fmaxf(pay1, lay1);
  const float rbx = fminf(pax2, lax2), rby = fminf(pay2, lay2);
  const float iw = fmaxf(rbx - ltx, 0.0f), ih = fmaxf(rby - lty, 0.0f);
  const float inter = iw * ih;
  const float areaA = (pax2 - pax1) * (pay2 - pay1);
  const float areaB = (lax2 - lax1) * (lay2 - lay1);
  const float iou = inter / (areaA + areaB - inter);
  const float prob = expf(cls[(size_t)b * 2 * NANCH + 2 * i + 1]);
  const bool valid = label_cls[(size_t)b * NANCH + i] > 0;

  __syncthreads();
  {
    unsigned long long bal = __ballot(valid);
    if (lane == 0) atomicAdd(&s_np, (int)__popcll(bal));
  }
  __syncthreads();
  const int np = s_np;

  // ---- pass 1: sort prob by descending (valid ? iou : NEG) ----------------
  s_key[tid] = valid ? iou : NEGK;
  s_val[tid] = prob;
  float tot1 = pair_loss_pass(s_key, s_val, s_u, s_v, s_usum, s_vsum,
                              s_part, s_wred, np, tid);

  // ---- pass 2: sort iou by descending (valid ? prob : NEG) ----------------
  s_key[tid] = valid ? prob : NEGK;
  s_val[tid] = iou;
  float tot2 = pair_loss_pass(s_key, s_val, s_u, s_v, s_usum, s_vsum,
                              s_part, s_wred, np, tid);

  if (tid == 0) {
    const float cnt = 0.5f * (float)np * (float)(np - 1);
    const float denom = fmaxf(cnt, 1.0f);
    const float ok = (np >= 2) ? 1.0f : 0.0f;
    ws[b * 3 + 0] = (tot1 / denom) * ok;
    ws[b * 3 + 1] = (tot2 / denom) * ok;
    ws[b * 3 + 2] = ok;
  }
}

__global__ void rank_igr_final(const float* __restrict__ ws,
                               float* __restrict__ out, int B) {
  // deterministic fixed-order reduction, single thread
  if (threadIdx.x == 0 && blockIdx.x == 0) {
    float s1 = 0.0f, s2 = 0.0f, cnt = 0.0f;
    for (int b = 0; b < B; ++b) {
      s1  += ws[b * 3 + 0];
      s2  += ws[b * 3 + 1];
      cnt += ws[b * 3 + 2];
    }
    const float nb = fmaxf(cnt, 1.0f);
    out[0] = s1 / nb;
    out[1] = s2 / nb;
  }
}

extern "C" void kernel_launch(void* const* d_in, const int* in_sizes, int n_in,
                              void* d_out, int out_size, void* d_ws, size_t ws_size,
                              hipStream_t stream) {
  const float* cls       = (const float*)d_in[0];  // (B, N, 2) f32
  const int*   label_cls = (const int*)d_in[1];    // (B, N)    i32
  const float* pred_loc  = (const float*)d_in[2];  // (B, 4, 32, 32) f32
  const float* label_loc = (const float*)d_in[3];  // (B, 4, 32, 32) f32
  const int B = in_sizes[1] / NANCH;               // label_cls flat = B*N
  float* ws = (float*)d_ws;                        // B * 3 floats

  rank_igr_main<<<B, NTHREADS, 0, stream>>>(cls, label_cls, pred_loc, label_loc, ws);
  rank_igr_final<<<1, 1, 0, stream>>>(ws, (float*)d_out, B);
}